// KeyActionRetrieval_489626271812
// MI455X (gfx1250) — compile-verified
//
#include <hip/hip_runtime.h>
#include <math.h>

typedef float v2f __attribute__((ext_vector_type(2)));
typedef float v8f __attribute__((ext_vector_type(8)));

#define D_DIM   2048
#define TOPK    8
#define ACT     7
#define WAVES   8
#define RPW     16              // rows per wave
#define RPB     (WAVES * RPW)   // 128 rows per block
#define KC      64              // K-chunk per iteration
#define TSTRIDE 68              // padded tile row stride (floats): 16B aligned, conflict-free

// ---------------------------------------------------------------------------
// Phase 1: sims[i] = (keys[i]·q) / max(||keys[i]|| * ||q||, eps)
// One wave computes 16 rows via V_WMMA_F32_16X16X4_F32; row norms ride along
// on the same global loads (zero extra HBM traffic).
// ---------------------------------------------------------------------------
__global__ __launch_bounds__(256)
void sims_kernel(const float* __restrict__ q, const float* __restrict__ keys,
                 float* __restrict__ sims, int N)
{
    __shared__ float qs[D_DIM];
    __shared__ float tile[WAVES][RPW * TSTRIDE];
    __shared__ float red[256];
    __shared__ float dotsL[WAVES][RPW];
    __shared__ float nsqL[WAVES][RPW];

    const int tid  = threadIdx.x;
    const int wave = tid >> 5;
    const int lane = tid & 31;
    const int half = lane >> 4;    // 0: lanes 0-15, 1: lanes 16-31
    const int l15  = lane & 15;

    // ---- stage query into LDS, compute ||q||^2 with a block tree-reduce ----
    float qp = 0.f;
    for (int i = tid; i < D_DIM; i += 256) {
        float v = q[i];
        qs[i] = v;
        qp = fmaf(v, v, qp);
    }
    red[tid] = qp;
    __syncthreads();
    for (int s = 128; s > 0; s >>= 1) {
        if (tid < s) red[tid] += red[tid + s];
        __syncthreads();
    }
    const float qnorm = sqrtf(red[0]);

    const long rowBase = (long)blockIdx.x * RPB + (long)wave * RPW;
    float* myTile = &tile[wave][0];

    v8f acc0 = {};
    v8f acc1 = {};
    float nsq[8];
#pragma unroll
    for (int j = 0; j < 8; ++j) nsq[j] = 0.f;

    for (int k0 = 0; k0 < D_DIM; k0 += KC) {
        // ---- global -> LDS: 16 rows x 64 floats, coalesced b128 per lane ----
        // round j: lanes 0-15 load row 2j (256 contiguous bytes),
        //          lanes 16-31 load row 2j+1.
#pragma unroll
        for (int j = 0; j < 8; ++j) {
            const int r = 2 * j + half;
            long grow = rowBase + r;
            if (grow >= N) grow = N - 1;            // clamp; result discarded
            const float* src = keys + grow * (long)D_DIM + k0 + l15 * 4;
            const float4 v = *(const float4*)src;
            __builtin_prefetch(src + KC, 0, 3);     // global_prefetch_b8, next chunk
            *(float4*)(myTile + r * TSTRIDE + l15 * 4) = v;
            // row sum-of-squares rides along for free
            nsq[j] = fmaf(v.x, v.x, nsq[j]);
            nsq[j] = fmaf(v.y, v.y, nsq[j]);
            nsq[j] = fmaf(v.z, v.z, nsq[j]);
            nsq[j] = fmaf(v.w, v.w, nsq[j]);
        }

        // ---- 16 WMMA steps of K=4 over the chunk (2 accs break RAW chain) --
        // A frag (16x4 f32): lane<16 -> {A[l15][k], A[l15][k+1]},
        //                    lane>=16 -> {A[l15][k+2], A[l15][k+3]}
        // B frag (4x16): query chunk broadcast across all 16 columns.
#pragma unroll
        for (int i = 0; i < 16; i += 2) {
            v2f a0 = *(const v2f*)(myTile + l15 * TSTRIDE + 4 * i + 2 * half);
            v2f b0 = *(const v2f*)(qs + k0 + 4 * i + 2 * half);
            acc0 = __builtin_amdgcn_wmma_f32_16x16x4_f32(
                false, a0, false, b0, (short)0, acc0, false, false);
            v2f a1 = *(const v2f*)(myTile + l15 * TSTRIDE + 4 * (i + 1) + 2 * half);
            v2f b1 = *(const v2f*)(qs + k0 + 4 * (i + 1) + 2 * half);
            acc1 = __builtin_amdgcn_wmma_f32_16x16x4_f32(
                false, a1, false, b1, (short)0, acc1, false, false);
        }
    }
    v8f acc = acc0 + acc1;

    // ---- reduce nsq across the 16 lanes of each half-wave ----
#pragma unroll
    for (int j = 0; j < 8; ++j) {
        float v = nsq[j];
        v += __shfl_xor(v, 1);
        v += __shfl_xor(v, 2);
        v += __shfl_xor(v, 4);
        v += __shfl_xor(v, 8);
        nsq[j] = v;
    }
    if (l15 == 0) {
#pragma unroll
        for (int j = 0; j < 8; ++j) nsqL[wave][2 * j + half] = nsq[j];
    }
    // D column 0: lane 0 holds rows 0..7, lane 16 holds rows 8..15
    if (lane == 0) {
#pragma unroll
        for (int r = 0; r < 8; ++r) dotsL[wave][r] = acc[r];
    }
    if (lane == 16) {
#pragma unroll
        for (int r = 0; r < 8; ++r) dotsL[wave][8 + r] = acc[r];
    }
    __syncthreads();

    if (lane < 16) {
        const long row = rowBase + lane;
        if (row < (long)N) {
            const float dot = dotsL[wave][lane];
            const float den = fmaxf(sqrtf(nsqL[wave][lane]) * qnorm, 1e-8f);
            sims[row] = dot / den;
        }
    }
}

// ---------------------------------------------------------------------------
// Phase 2: top-8 of sims[N] + gather actions -> out[8][7]
// ---------------------------------------------------------------------------
__device__ __forceinline__ void ins8(float v, int i, float tv[TOPK], int ti[TOPK])
{
    if (v > tv[TOPK - 1]) {
        tv[TOPK - 1] = v;
        ti[TOPK - 1] = i;
#pragma unroll
        for (int j = TOPK - 1; j > 0; --j) {
            if (tv[j] > tv[j - 1]) {
                float fv = tv[j]; tv[j] = tv[j - 1]; tv[j - 1] = fv;
                int   fi = ti[j]; ti[j] = ti[j - 1]; ti[j - 1] = fi;
            }
        }
    }
}

__global__ __launch_bounds__(1024)
void topk_kernel(const float* __restrict__ sims, const float* __restrict__ actions,
                 float* __restrict__ out, int N)
{
    __shared__ float wv[32][TOPK];
    __shared__ int   wi[32][TOPK];

    float tv[TOPK];
    int   ti[TOPK];
#pragma unroll
    for (int j = 0; j < TOPK; ++j) { tv[j] = -INFINITY; ti[j] = 0; }

    const int tid  = threadIdx.x;
    const int lane = tid & 31;
    const int warp = tid >> 5;

    for (int i = tid; i < N; i += 1024) ins8(sims[i], i, tv, ti);

    // wave32 shuffle-merge of sorted-8 lists
#pragma unroll
    for (int off = 16; off >= 1; off >>= 1) {
        float ov[TOPK]; int oi[TOPK];
#pragma unroll
        for (int j = 0; j < TOPK; ++j) {
            ov[j] = __shfl_down(tv[j], off);
            oi[j] = __shfl_down(ti[j], off);
        }
#pragma unroll
        for (int j = 0; j < TOPK; ++j) ins8(ov[j], oi[j], tv, ti);
    }
    if (lane == 0) {
#pragma unroll
        for (int j = 0; j < TOPK; ++j) { wv[warp][j] = tv[j]; wi[warp][j] = ti[j]; }
    }
    __syncthreads();

    if (warp == 0) {
#pragma unroll
        for (int j = 0; j < TOPK; ++j) { tv[j] = wv[lane][j]; ti[j] = wi[lane][j]; }
#pragma unroll
        for (int off = 16; off >= 1; off >>= 1) {
            float ov[TOPK]; int oi[TOPK];
#pragma unroll
            for (int j = 0; j < TOPK; ++j) {
                ov[j] = __shfl_down(tv[j], off);
                oi[j] = __shfl_down(ti[j], off);
            }
#pragma unroll
            for (int j = 0; j < TOPK; ++j) ins8(ov[j], oi[j], tv, ti);
        }
        if (lane == 0) {
            for (int k = 0; k < TOPK; ++k) {
                const long idx = (ti[k] < 0) ? 0 : (long)ti[k];
                for (int a = 0; a < ACT; ++a)
                    out[k * ACT + a] = actions[idx * ACT + a];
            }
        }
    }
}

// ---------------------------------------------------------------------------
extern "C" void kernel_launch(void* const* d_in, const int* in_sizes, int n_in,
                              void* d_out, int out_size, void* d_ws, size_t ws_size,
                              hipStream_t stream)
{
    const float* q       = (const float*)d_in[0];
    const float* keys    = (const float*)d_in[1];
    const float* actions = (const float*)d_in[2];
    const int D = in_sizes[0];          // 2048 (kernel hardcodes D_DIM)
    const int N = in_sizes[1] / D;      // 100000

    float* sims = (float*)d_ws;         // N floats of scratch

    const int blocks = (N + RPB - 1) / RPB;
    sims_kernel<<<blocks, 256, 0, stream>>>(q, keys, sims, N);
    topk_kernel<<<1, 1024, 0, stream>>>(sims, actions, (float*)d_out, N);

    (void)n_in; (void)out_size; (void)ws_size;
}